// I2E_42107859370402
// MI455X (gfx1250) — compile-verified
//
#include <hip/hip_runtime.h>
#include <stdint.h>

#ifndef __has_builtin
#define __has_builtin(x) 0
#endif

#if __has_builtin(__builtin_amdgcn_global_load_async_to_lds_b32)
#define HAVE_ASYNC_LDS 1
#else
#define HAVE_ASYNC_LDS 0
#pragma message("gfx1250 async-to-LDS builtin NOT found; using synchronous LDS staging fallback")
#endif

typedef float v4f __attribute__((ext_vector_type(4)));
typedef float v2f __attribute__((ext_vector_type(2)));

#define BATCH 16
#define CHN   3
#define HH    512
#define WW    512
#define HWSZ  (HH * WW)
#define RATIO_F 0.12f

// Stencil tile: 128 wide x 8 tall outputs per 256-thread block (32x8, 4 px/thread)
#define TW    128
#define TH    8
#define TCOLS (TW + 2)
#define TROWS (TH + 2)
#define TSTR  132   // padded LDS row stride (multiple of 4 for b128 LDS reads)

__device__ __forceinline__ int clampi(int v, int lo, int hi) {
  return v < lo ? lo : (v > hi ? hi : v);
}

// ---------------------------------------------------------------------------
// Kernel 0: init per-image min/max slots (uint bit patterns; inputs are >= 0
// so unsigned-int order == float order).  ws[2b]=min, ws[2b+1]=max.
// ---------------------------------------------------------------------------
__global__ void mm_init_kernel(unsigned* __restrict__ mm) {
  int i = threadIdx.x;
  if (i < BATCH) {
    mm[2 * i]     = 0x7F7FFFFFu;  // +FLT_MAX bits
    mm[2 * i + 1] = 0u;           // +0.0f bits
  }
}

// ---------------------------------------------------------------------------
// Kernel 1: channel-max + per-image min/max reduction (reads 48 MB, warms L2)
// grid = BATCH*256 blocks of 256 threads, 4 pixels (one float4) per thread.
// ---------------------------------------------------------------------------
__global__ void __launch_bounds__(256)
chanmax_minmax_kernel(const float* __restrict__ img, unsigned* __restrict__ mm) {
  const int b   = blockIdx.x >> 8;
  const int blk = blockIdx.x & 255;
  const int tid = threadIdx.x;
  const size_t base = (size_t)b * (CHN * HWSZ) + (size_t)blk * 1024 + (size_t)tid * 4;

  __builtin_prefetch(img + base + 1024);  // -> global_prefetch_b8

  v4f a = *(const v4f*)(img + base);
  v4f c = *(const v4f*)(img + base + HWSZ);
  v4f d = *(const v4f*)(img + base + 2 * HWSZ);
  v4f v;
  v.x = fmaxf(a.x, fmaxf(c.x, d.x));
  v.y = fmaxf(a.y, fmaxf(c.y, d.y));
  v.z = fmaxf(a.z, fmaxf(c.z, d.z));
  v.w = fmaxf(a.w, fmaxf(c.w, d.w));

  float mn = fminf(fminf(v.x, v.y), fminf(v.z, v.w));
  float mx = fmaxf(fmaxf(v.x, v.y), fmaxf(v.z, v.w));

  __shared__ float smin[256];
  __shared__ float smax[256];
  smin[tid] = mn;
  smax[tid] = mx;
  __syncthreads();
  for (int s = 128; s > 0; s >>= 1) {
    if (tid < s) {
      smin[tid] = fminf(smin[tid], smin[tid + s]);
      smax[tid] = fmaxf(smax[tid], smax[tid + s]);
    }
    __syncthreads();
  }
  if (tid == 0) {
    atomicMin(&mm[2 * b],     __float_as_uint(smin[0]));
    atomicMax(&mm[2 * b + 1], __float_as_uint(smax[0]));
  }
}

// ---------------------------------------------------------------------------
// Kernel 2: 3x3 neighbor-difference stencil + threshold, 16 output planes.
// Stages channel tiles into LDS with gfx1250 async global->LDS loads.
// ---------------------------------------------------------------------------
__global__ void __launch_bounds__(256)
edge_threshold_kernel(const float* __restrict__ img,
                      const unsigned* __restrict__ mm,
                      float* __restrict__ out) {
  // (pos_x, pos_y, neg_x, neg_y) per filter; padded[p + h] - padded[n + w] form
  constexpr int PX[8] = {1, 0, 2, 0, 1, 0, 0, 0};
  constexpr int PY[8] = {0, 2, 1, 0, 2, 1, 2, 0};
  constexpr int NX[8] = {2, 1, 0, 2, 1, 1, 1, 1};
  constexpr int NY[8] = {2, 0, 2, 1, 1, 1, 1, 1};

  const int tx  = threadIdx.x;            // 0..31
  const int ty  = threadIdx.y;            // 0..7
  const int tid = ty * 32 + tx;
  const int w0  = blockIdx.x * TW;
  const int h0  = blockIdx.y * TH;
  const int b   = blockIdx.z;

  __shared__ float smem_v[TROWS * TSTR];  // channel-max tile incl. halo

#if HAVE_ASYNC_LDS
  __shared__ float smem_ch[3 * TROWS * TSTR];
  // Async-stage the three raw channel tiles (with replicate-clamped halo).
  for (int i = tid; i < 3 * TROWS * TCOLS; i += 256) {
    const int ch  = i / (TROWS * TCOLS);
    const int rem = i - ch * (TROWS * TCOLS);
    const int r   = rem / TCOLS;
    const int c   = rem - r * TCOLS;
    const int gy  = clampi(h0 - 1 + r, 0, HH - 1);
    const int gx  = clampi(w0 - 1 + c, 0, WW - 1);
    const float* g = img + ((size_t)(b * CHN + ch) * HWSZ + (size_t)gy * WW + gx);
    const float* l = &smem_ch[ch * (TROWS * TSTR) + r * TSTR + c];
    __builtin_amdgcn_global_load_async_to_lds_b32(
        (__attribute__((address_space(1))) int*)(uintptr_t)g,
        (__attribute__((address_space(3))) int*)(uint32_t)(uintptr_t)l,
        0, 0);
  }
#if __has_builtin(__builtin_amdgcn_s_wait_asynccnt)
  __builtin_amdgcn_s_wait_asynccnt(0);
#else
  asm volatile("s_wait_asynccnt 0" ::: "memory");
#endif
  __syncthreads();
  // Reduce the three channel tiles to the channel-max tile.
  for (int i = tid; i < TROWS * TCOLS; i += 256) {
    const int r   = i / TCOLS;
    const int c   = i - r * TCOLS;
    const int idx = r * TSTR + c;
    smem_v[idx] = fmaxf(smem_ch[idx],
                  fmaxf(smem_ch[TROWS * TSTR + idx], smem_ch[2 * TROWS * TSTR + idx]));
  }
#else
  // Synchronous fallback: load 3 channels, reduce, store to LDS.
  for (int i = tid; i < TROWS * TCOLS; i += 256) {
    const int r  = i / TCOLS;
    const int c  = i - r * TCOLS;
    const int gy = clampi(h0 - 1 + r, 0, HH - 1);
    const int gx = clampi(w0 - 1 + c, 0, WW - 1);
    const size_t gbase = (size_t)b * (CHN * HWSZ) + (size_t)gy * WW + gx;
    smem_v[r * TSTR + c] =
        fmaxf(img[gbase], fmaxf(img[gbase + HWSZ], img[gbase + 2 * HWSZ]));
  }
#endif
  __syncthreads();

  const float thr =
      (__uint_as_float(mm[2 * b + 1]) - __uint_as_float(mm[2 * b])) * RATIO_F;

  // Register patch: 3 rows x 6 cols covers the 3x3 windows of 4 adjacent pixels.
  const int oy  = ty;
  const int ox0 = tx * 4;
  float arr[3][6];
#pragma unroll
  for (int dy = 0; dy < 3; ++dy) {
    const float* rp = &smem_v[(oy + dy) * TSTR + ox0];  // 16B aligned
    v4f h4 = *(const v4f*)rp;
    v2f h2 = *(const v2f*)(rp + 4);
    arr[dy][0] = h4.x; arr[dy][1] = h4.y; arr[dy][2] = h4.z; arr[dy][3] = h4.w;
    arr[dy][4] = h2.x; arr[dy][5] = h2.y;
  }

  float resp[8][4], resn[8][4];
#pragma unroll
  for (int p = 0; p < 4; ++p) {
#pragma unroll
    for (int k = 0; k < 8; ++k) {
      const float dd  = arr[PX[k]][p + PY[k]] - arr[NX[k]][p + NY[k]];
      const float pos = fmaxf(dd, 0.0f);
      const float neg = fmaxf(-dd, 0.0f);
      resp[k][p] = (pos >= thr) ? 1.0f : 0.0f;
      resn[k][p] = (neg >= thr) ? 1.0f : 0.0f;
    }
  }

  // 16 non-temporal b128 stores per thread: plane (k*16+b)*2+s = k*32 + 2b + s.
  const size_t pixoff = (size_t)(h0 + oy) * WW + (size_t)(w0 + ox0);
#pragma unroll
  for (int k = 0; k < 8; ++k) {
    v4f vp; vp.x = resp[k][0]; vp.y = resp[k][1]; vp.z = resp[k][2]; vp.w = resp[k][3];
    v4f vn; vn.x = resn[k][0]; vn.y = resn[k][1]; vn.z = resn[k][2]; vn.w = resn[k][3];
    const size_t plane = (size_t)(k * (2 * BATCH) + 2 * b);
    __builtin_nontemporal_store(vp, (v4f*)(out + plane * HWSZ + pixoff));
    __builtin_nontemporal_store(vn, (v4f*)(out + (plane + 1) * HWSZ + pixoff));
  }
}

// ---------------------------------------------------------------------------
extern "C" void kernel_launch(void* const* d_in, const int* in_sizes, int n_in,
                              void* d_out, int out_size, void* d_ws, size_t ws_size,
                              hipStream_t stream) {
  (void)in_sizes; (void)n_in; (void)out_size; (void)ws_size;
  const float* img = (const float*)d_in[0];
  float* out  = (float*)d_out;
  unsigned* mm = (unsigned*)d_ws;  // 32 uints: per-image (min,max) bit patterns

  mm_init_kernel<<<1, 32, 0, stream>>>(mm);
  chanmax_minmax_kernel<<<BATCH * 256, 256, 0, stream>>>(img, mm);

  dim3 grid(WW / TW, HH / TH, BATCH);  // (4, 64, 16)
  dim3 block(32, 8, 1);
  edge_threshold_kernel<<<grid, block, 0, stream>>>(img, mm, out);
}